// Supported_CantileverPINN_tuned_53094385713965
// MI455X (gfx1250) — compile-verified
//
#include <hip/hip_runtime.h>
#include <hip/hip_bf16.h>

typedef __attribute__((ext_vector_type(2))) float v2f;
typedef __attribute__((ext_vector_type(8))) float v8f;

#define HID 15        // real hidden width
#define D   16        // padded hidden width
#define WAVES 8       // waves per block in the field kernel

// ---------------------------------------------------------------------------
// Branch-free tanh: prefer gfx1250 hardware v_tanh_f32, else exp2-based.
// ---------------------------------------------------------------------------
__device__ __forceinline__ float fast_tanh(float u)
{
#if __has_builtin(__builtin_amdgcn_tanhf)
  return __builtin_amdgcn_tanhf(u);
#else
  float au = __builtin_fabsf(u);
#if __has_builtin(__builtin_amdgcn_exp2f)
  float t = __builtin_amdgcn_exp2f(-2.885390081777927f * au);  // e^{-2|u|}
#else
  float t = exp2f(-2.885390081777927f * au);
#endif
#if __has_builtin(__builtin_amdgcn_rcpf)
  float r = __builtin_amdgcn_rcpf(1.f + t);
#else
  float r = 1.f / (1.f + t);
#endif
  float s = (1.f - t) * r;
  return __builtin_copysignf(s, u);
#endif
}

// ---------------------------------------------------------------------------
// Taylor-jet composition through tanh (Faa di Bruno, order 4).
// tanh' = 1-s^2 ; tanh'' = -2 s (1-s^2) ; tanh''' = -2(1-s^2)(1-3s^2) ;
// tanh'''' = 8 s (1-s^2)(2-3s^2)
// ---------------------------------------------------------------------------
__device__ __forceinline__ void jet_tanh(float u, float u1, float u2, float u3, float u4,
                                         float& y0, float& y1, float& y2, float& y3, float& y4)
{
  float s  = fast_tanh(u);
  float s2 = s * s;
  float t1 = 1.f - s2;
  float t2 = -2.f * s * t1;
  float t3 = -2.f * t1 * (1.f - 3.f * s2);
  float t4 = 8.f  * s  * t1 * (2.f - 3.f * s2);
  float u1s = u1 * u1;
  y0 = s;
  y1 = t1 * u1;
  y2 = t2 * u1s + t1 * u2;
  y3 = t3 * u1s * u1 + 3.f * t2 * u1 * u2 + t1 * u3;
  y4 = t4 * u1s * u1s + 6.f * t3 * u1s * u2
     + t2 * (3.f * u2 * u2 + 4.f * u1 * u3) + t1 * u4;
}

// ---------------------------------------------------------------------------
// Load one 16x16 B operand (weights, col-major K-adjacent in LDS) into the
// 4 K-chunk register pairs used by V_WMMA_F32_16X16X4_F32.
// B chunk layout (4x16 f32): vgpr0 = K {0 lo-lanes | 2 hi-lanes}, vgpr1 = K {1|3}
// ---------------------------------------------------------------------------
__device__ __forceinline__ void load_B(const float* __restrict__ sWT, int lane,
                                       v2f (&B)[4])
{
  const int row = lane & 15;          // N index
  const int hi2 = (lane >> 4) << 1;   // per-half K sub-offset (0 or 2)
#pragma unroll
  for (int c = 0; c < 4; ++c) {
    const int kk = 4 * c + hi2;
    B[c].x = sWT[row * D + kk];
    B[c].y = sWT[row * D + kk + 1];
  }
}

// ---------------------------------------------------------------------------
// 16x16x16 f32 GEMM as 4 chained V_WMMA_F32_16X16X4_F32 (B held in registers).
// sA: 16x16 row-major [m*D + k] (A-matrix, M=points, K=units_in)
// A chunk layout (16x4 f32): vgpr0 = K {0 lo-lanes | 2 hi-lanes}, vgpr1 = K {1|3}
// ---------------------------------------------------------------------------
__device__ __forceinline__ v8f wmma_16x16x16(const float* __restrict__ sA,
                                             const v2f (&B)[4], int lane)
{
  const int row = lane & 15;
  const int hi2 = (lane >> 4) << 1;
  v8f acc = {0.f,0.f,0.f,0.f,0.f,0.f,0.f,0.f};
#pragma unroll
  for (int c = 0; c < 4; ++c) {
    const int kk = 4 * c + hi2;
    v2f a;
    a.x = sA[row * D + kk];
    a.y = sA[row * D + kk + 1];
    acc = __builtin_amdgcn_wmma_f32_16x16x4_f32(false, a, false, B[c], (short)0, acc,
                                                false, false);
  }
  return acc;
}

// ---------------------------------------------------------------------------
// Cooperative weight staging into LDS (zero-padded 15 -> 16, W2/W3 transposed
// so the K (contraction) index is unit-stride for WMMA operand loads).
// ---------------------------------------------------------------------------
__device__ __forceinline__ void load_weights_lds(
    const float* __restrict__ W1, const float* __restrict__ b1,
    const float* __restrict__ W2, const float* __restrict__ b2,
    const float* __restrict__ W3, const float* __restrict__ b3,
    const float* __restrict__ W4,
    float* sW1, float* sB1, float* sW2T, float* sB2,
    float* sW3T, float* sB3, float* sW4)
{
  for (int i = threadIdx.x; i < D * D; i += blockDim.x) {
    int k = i >> 4, n = i & 15;
    bool in = (k < HID) && (n < HID);
    sW2T[n * D + k] = in ? W2[k * HID + n] : 0.f;
    sW3T[n * D + k] = in ? W3[k * HID + n] : 0.f;
  }
  for (int i = threadIdx.x; i < D; i += blockDim.x) {
    bool in = (i < HID);
    sW1[i] = in ? W1[i] : 0.f;
    sB1[i] = in ? b1[i] : 0.f;
    sB2[i] = in ? b2[i] : 0.f;
    sB3[i] = in ? b3[i] : 0.f;
    sW4[i] = in ? W4[i] : 0.f;
  }
}

// ---------------------------------------------------------------------------
// Propagate the full 5-component jet for a 16-point tile through the MLP up
// to (and including) the layer-3 tanh.  Result in C-matrix register layout:
// h?[r] @ lane L  ==  element (point m = r + 8*(L>>4), unit j = L&15).
// `active` must be wave-uniform; barriers are executed unconditionally.
// ---------------------------------------------------------------------------
__device__ __forceinline__ void mlp_jets_tile(
    bool active, float xv, int lane,
    const float* sW1, const float* sB1, const float* sB2, const float* sB3,
    const v2f (&B2)[4], const v2f (&B3)[4],
    float* sHw,                               // this wave's 5*256-float scratch
    float (&h0)[8], float (&h1)[8], float (&h2)[8], float (&h3)[8], float (&h4)[8])
{
  const int hi = lane >> 4;
  const int j  = lane & 15;
  const float w1j = sW1[j], b1j = sB1[j];

  if (active) {
#pragma unroll
    for (int r = 0; r < 8; ++r) {
      float xm = __shfl(xv, r + 8 * hi, 32);  // x of point m = r + 8*hi
      // inner jet of (x*W1_j + b1_j): u1 = W1_j, u2..u4 = 0
      jet_tanh(xm * w1j + b1j, w1j, 0.f, 0.f, 0.f,
               h0[r], h1[r], h2[r], h3[r], h4[r]);
    }
  }

#pragma unroll
  for (int l = 0; l < 2; ++l) {
    const float bj = l ? sB3[j] : sB2[j];

    __syncthreads();
    if (active) {
#pragma unroll
      for (int r = 0; r < 8; ++r) {
        int idx = (r + 8 * hi) * D + j;       // row-major [m][k] staging
        sHw[0 * 256 + idx] = h0[r];
        sHw[1 * 256 + idx] = h1[r];
        sHw[2 * 256 + idx] = h2[r];
        sHw[3 * 256 + idx] = h3[r];
        sHw[4 * 256 + idx] = h4[r];
      }
    }
    __syncthreads();

    if (active) {
      v8f a0, a1, a2, a3, a4;
      if (l == 0) {
        a0 = wmma_16x16x16(sHw + 0 * 256, B2, lane);
        a1 = wmma_16x16x16(sHw + 1 * 256, B2, lane);
        a2 = wmma_16x16x16(sHw + 2 * 256, B2, lane);
        a3 = wmma_16x16x16(sHw + 3 * 256, B2, lane);
        a4 = wmma_16x16x16(sHw + 4 * 256, B2, lane);
      } else {
        a0 = wmma_16x16x16(sHw + 0 * 256, B3, lane);
        a1 = wmma_16x16x16(sHw + 1 * 256, B3, lane);
        a2 = wmma_16x16x16(sHw + 2 * 256, B3, lane);
        a3 = wmma_16x16x16(sHw + 3 * 256, B3, lane);
        a4 = wmma_16x16x16(sHw + 4 * 256, B3, lane);
      }
#pragma unroll
      for (int r = 0; r < 8; ++r)
        jet_tanh(a0[r] + bj, a1[r], a2[r], a3[r], a4[r],
                 h0[r], h1[r], h2[r], h3[r], h4[r]);
    }
  }
}

// ---------------------------------------------------------------------------
// Kernel A: boundary jets at x=0 (point m=0) and x=L=1 (point m=1).
// ws[0]=w(0) ws[1]=w_x(0) ws[2]=w(L) ws[3]=w_xx(L) ws[4]=0 (pde accumulator)
// ---------------------------------------------------------------------------
__global__ void __launch_bounds__(32)
pinn_bc_kernel(const float* __restrict__ W1, const float* __restrict__ b1,
               const float* __restrict__ W2, const float* __restrict__ b2,
               const float* __restrict__ W3, const float* __restrict__ b3,
               const float* __restrict__ W4, const float* __restrict__ b4,
               float* __restrict__ ws)
{
  __shared__ float sW1[D], sB1[D], sB2[D], sB3[D], sW4[D];
  __shared__ float sW2T[D * D], sW3T[D * D];
  __shared__ float sH[5 * 256];

  load_weights_lds(W1, b1, W2, b2, W3, b3, W4, sW1, sB1, sW2T, sB2, sW3T, sB3, sW4);
  __syncthreads();

  const int lane = threadIdx.x;
  const int j    = lane & 15;
  const float xv = (j == 1) ? 1.0f : 0.0f;    // m=0 -> x=0 ; m=1 -> x=LENGTH

  v2f B2r[4], B3r[4];
  load_B(sW2T, lane, B2r);
  load_B(sW3T, lane, B3r);

  float h0[8], h1[8], h2[8], h3[8], h4[8];
  mlp_jets_tile(true, xv, lane, sW1, sB1, sB2, sB3, B2r, B3r, sH,
                h0, h1, h2, h3, h4);

  const float w4j = sW4[j];
  const float b4v = b4[0];
#pragma unroll
  for (int r = 0; r < 2; ++r) {               // low half: r==m for m=0,1
    float p0 = h0[r] * w4j;
    float p1 = h1[r] * w4j;
    float p2 = h2[r] * w4j;
#pragma unroll
    for (int off = 1; off < 16; off <<= 1) {
      p0 += __shfl_xor(p0, off, 16);
      p1 += __shfl_xor(p1, off, 16);
      p2 += __shfl_xor(p2, off, 16);
    }
    if (lane == 0) {
      if (r == 0) { ws[0] = p0 + b4v; ws[1] = p1; }
      else        { ws[2] = p0 + b4v; ws[3] = p2; }
    }
  }
  if (lane == 0) ws[4] = 0.f;                 // zero pde accumulator each call
}

// ---------------------------------------------------------------------------
// Kernel B: field pass — per point compute w_xxxx, accumulate sum((w_xxxx+1)^2)
// ---------------------------------------------------------------------------
__global__ void __launch_bounds__(32 * WAVES)
pinn_field_kernel(const float* __restrict__ X,
                  const float* __restrict__ W1, const float* __restrict__ b1,
                  const float* __restrict__ W2, const float* __restrict__ b2,
                  const float* __restrict__ W3, const float* __restrict__ b3,
                  const float* __restrict__ W4,
                  float* __restrict__ ws_accum, int n)
{
  __shared__ float sW1[D], sB1[D], sB2[D], sB3[D], sW4[D];
  __shared__ float sW2T[D * D], sW3T[D * D];
  __shared__ float sH[WAVES * 5 * 256];
  __shared__ float sRed[32 * WAVES];

  load_weights_lds(W1, b1, W2, b2, W3, b3, W4, sW1, sB1, sW2T, sB2, sW3T, sB3, sW4);
  __syncthreads();

  const int lane = threadIdx.x & 31;
  const int wave = threadIdx.x >> 5;
  const int hi   = lane >> 4;
  const int j    = lane & 15;
  float* sHw = sH + wave * 5 * 256;
  const float w4j = sW4[j];

  // Weight B-operands are loop-invariant: keep them in registers for the
  // whole kernel (staged LDS weights are never overwritten).
  v2f B2r[4], B3r[4];
  load_B(sW2T, lane, B2r);
  load_B(sW3T, lane, B3r);

  const int nt = (n + 15) >> 4;               // 16-point tiles
  float local = 0.f;

  // Block-uniform trip count (barriers inside mlp_jets_tile stay convergent).
  for (int base = blockIdx.x * WAVES; base < nt; base += gridDim.x * WAVES) {
    const int  t      = base + wave;
    const bool active = (t < nt);
    const int  g      = t << 4;

    float xv = 0.f;
    if (active) {
      int xi = g + j;
      xv = (xi < n) ? X[xi] : 0.f;
    }

    float h0[8], h1[8], h2[8], h3[8], h4[8];
    mlp_jets_tile(active, xv, lane, sW1, sB1, sB2, sB3, B2r, B3r, sHw,
                  h0, h1, h2, h3, h4);

    if (active) {
#pragma unroll
      for (int r = 0; r < 8; ++r) {
        float pk = h4[r] * w4j;               // final layer, 4th-deriv jet only
        pk += __shfl_xor(pk, 1, 16);
        pk += __shfl_xor(pk, 2, 16);
        pk += __shfl_xor(pk, 4, 16);
        pk += __shfl_xor(pk, 8, 16);
        float res = pk + 1.0f;                // w_xxxx + P/(E*I), P=E=I=1
        int pidx = g + r + 8 * hi;
        if (j == 0 && pidx < n) local += res * res;
      }
    }
  }

  sRed[threadIdx.x] = local;
  __syncthreads();
  for (int s = (32 * WAVES) >> 1; s > 0; s >>= 1) {
    if (threadIdx.x < (unsigned)s) sRed[threadIdx.x] += sRed[threadIdx.x + s];
    __syncthreads();
  }
  if (threadIdx.x == 0) atomicAdd(ws_accum, sRed[0]);
}

// ---------------------------------------------------------------------------
// Kernel C: broadcast-fill the 5 output arrays + pde scalar.
// Layout: [loss(n)] [pde(1)] [w0^2(n)] [w0x^2(n)] [mL^2(n)] [wL^2(n)]
// ---------------------------------------------------------------------------
__global__ void pinn_fill_kernel(const float* __restrict__ ws,
                                 float* __restrict__ out, int n, float inv_n)
{
  const float pde   = ws[4] * inv_n;
  const float l_w0  = ws[0] * ws[0];
  const float l_w0x = ws[1] * ws[1];
  const float l_mL  = ws[3] * ws[3];
  const float l_wL  = ws[2] * ws[2];
  const float loss  = pde + l_w0 + l_w0x + l_mL + l_wL;

  int i = blockIdx.x * blockDim.x + threadIdx.x;
  if (i == 0) out[n] = pde;
  for (; i < n; i += gridDim.x * blockDim.x) {
    out[i]             = loss;
    out[n + 1 + i]     = l_w0;
    out[2 * n + 1 + i] = l_w0x;
    out[3 * n + 1 + i] = l_mL;
    out[4 * n + 1 + i] = l_wL;
  }
}

// ---------------------------------------------------------------------------
extern "C" void kernel_launch(void* const* d_in, const int* in_sizes, int n_in,
                              void* d_out, int out_size, void* d_ws, size_t ws_size,
                              hipStream_t stream)
{
  (void)n_in; (void)out_size; (void)ws_size;
  const float* X  = (const float*)d_in[0];
  const float* W1 = (const float*)d_in[1];
  const float* b1 = (const float*)d_in[2];
  const float* W2 = (const float*)d_in[3];
  const float* b2 = (const float*)d_in[4];
  const float* W3 = (const float*)d_in[5];
  const float* b3 = (const float*)d_in[6];
  const float* W4 = (const float*)d_in[7];
  const float* b4 = (const float*)d_in[8];
  const int n = in_sizes[0];

  float* ws  = (float*)d_ws;
  float* out = (float*)d_out;

  pinn_bc_kernel<<<1, 32, 0, stream>>>(W1, b1, W2, b2, W3, b3, W4, b4, ws);

  int nt = (n + 15) / 16;
  int blocks = (nt + WAVES - 1) / WAVES;
  if (blocks > 1024) blocks = 1024;
  if (blocks < 1) blocks = 1;
  pinn_field_kernel<<<blocks, 32 * WAVES, 0, stream>>>(
      X, W1, b1, W2, b2, W3, b3, W4, ws + 4, n);

  int fblocks = (n + 255) / 256;
  if (fblocks > 4096) fblocks = 4096;
  if (fblocks < 1) fblocks = 1;
  pinn_fill_kernel<<<fblocks, 256, 0, stream>>>(ws, out, n, 1.0f / (float)n);
}